// AssociativeMemory_78340203479338
// MI455X (gfx1250) — compile-verified
//
#include <hip/hip_runtime.h>
#include <hip/hip_bf16.h>
#include <math.h>

typedef __attribute__((ext_vector_type(2))) float v2f;
typedef __attribute__((ext_vector_type(8))) float v8f;

constexpr int B = 8, N = 2048, C = 256, R = 4;
constexpr float EPS = 1e-6f;

// ---- workspace layout (float offsets) ----
constexpr size_t U_OFF      = 0;                               // [B*N] usage u
constexpr size_t LOGW_OFF   = U_OFF      + (size_t)B * N;      // [B*N] write logits
constexpr size_t ALLOC_OFF  = LOGW_OFF   + (size_t)B * N;      // [B*N] allocation
constexpr size_t WW_OFF     = ALLOC_OFF  + (size_t)B * N;      // [B*N] write weighting
constexpr size_t RWSTK_OFF  = WW_OFF     + (size_t)B * N;      // [B*N*16] cols 0-3 rw, 4-7 (1-ww)*rw, 8-15 zero
constexpr size_t LOGR_OFF   = RWSTK_OFF  + (size_t)B * N * 16; // [B*N*R] read logits -> cr
constexpr size_t FWD2_OFF   = LOGR_OFF   + (size_t)B * N * R;  // [B*N*8] L@rw | L@rw1
constexpr size_t BWD2_OFF   = FWD2_OFF   + (size_t)B * N * 8;  // [B*N*8] Lt@rw | Lt@rw1
constexpr size_t RWNEW_OFF  = BWD2_OFF   + (size_t)B * N * 8;  // [B*N*R]
constexpr size_t MEMNEW_OFF = RWNEW_OFF  + (size_t)B * N * R;  // [B*N*C]
constexpr size_t SCAL_OFF   = MEMNEW_OFF + (size_t)B * N * C;  // scalars below
constexpr int WKN_S = 0;    // [B]   ||write_key||
constexpr int RKN_S = 8;    // [B*R] ||read_key||
constexpr int SPR_S = 40;   // [B*R] sum prec*rw
constexpr int SWR_S = 72;   // [B*R] sum ww*rw

// ---------- block reduction helpers ----------
__device__ __forceinline__ float blk_sum(float* red, int t, int nt, float v) {
  red[t] = v; __syncthreads();
  for (int s = nt >> 1; s > 0; s >>= 1) { if (t < s) red[t] += red[t + s]; __syncthreads(); }
  float r = red[0]; __syncthreads(); return r;
}
__device__ __forceinline__ float blk_max(float* red, int t, int nt, float v) {
  red[t] = v; __syncthreads();
  for (int s = nt >> 1; s > 0; s >>= 1) { if (t < s) red[t] = fmaxf(red[t], red[t + s]); __syncthreads(); }
  float r = red[0]; __syncthreads(); return r;
}

// ---------- K0: per-batch norms and Spr ----------
__global__ void k0_scalars(const float* __restrict__ wk, const float* __restrict__ rk,
                           const float* __restrict__ prec, const float* __restrict__ rw_in,
                           float* __restrict__ ws) {
  __shared__ float red[256];
  int b = blockIdx.x, t = threadIdx.x;
  float v = wk[b * C + t];
  float s = blk_sum(red, t, 256, v * v);
  if (t == 0) ws[SCAL_OFF + WKN_S + b] = sqrtf(s);
  for (int r = 0; r < R; ++r) {
    float u = rk[(size_t)b * C * R + (size_t)t * R + r];
    float ss = blk_sum(red, t, 256, u * u);
    if (t == 0) ws[SCAL_OFF + RKN_S + b * R + r] = sqrtf(ss);
  }
  for (int r = 0; r < R; ++r) {
    float acc = 0.f;
    for (int n = t; n < N; n += 256)
      acc += prec[(size_t)b * N + n] * rw_in[((size_t)b * N + n) * R + r];
    float ss = blk_sum(red, t, 256, acc);
    if (t == 0) ws[SCAL_OFF + SPR_S + b * R + r] = ss;
  }
}

// ---------- K1: usage u and write-content logits (wave per memory row) ----------
__global__ void k1_usage_phiw(const float* __restrict__ mem, const float* __restrict__ usage,
                              const float* __restrict__ rw_in, const float* __restrict__ wwp,
                              const float* __restrict__ fg, const float* __restrict__ wk,
                              const float* __restrict__ wstr, float* __restrict__ ws) {
  int wid = threadIdx.x >> 5, lane = threadIdx.x & 31;
  int g = blockIdx.x * 8 + wid;            // flat row b*N+n
  int b = g >> 11;
  float psi = 1.f;
  for (int r = 0; r < R; ++r)
    psi *= 1.f - fg[b * R + r] * rw_in[(size_t)g * R + r];
  float us = usage[g], wp = wwp[g];
  float uval = (us + wp - us * wp) * psi;
  size_t row = (size_t)g * C;
  float dot = 0.f, nrm = 0.f;
  for (int it = 0; it < C / 32; ++it) {
    int c = lane + 32 * it;
    float m = mem[row + c];
    dot += m * wk[b * C + c];
    nrm += m * m;
  }
  for (int off = 16; off > 0; off >>= 1) {
    dot += __shfl_xor(dot, off, 32);
    nrm += __shfl_xor(nrm, off, 32);
  }
  if (lane == 0) {
    ws[U_OFF + g] = uval;
    float kn = ws[SCAL_OFF + WKN_S + b];
    ws[LOGW_OFF + g] = dot / (sqrtf(nrm) * kn + EPS) * wstr[b];
  }
}

// ---------- K2: per-batch bitonic sort + multiplicative scan -> allocation ----------
__global__ __launch_bounds__(1024) void k2_sort_alloc(float* __restrict__ ws) {
  __shared__ float key[N];
  __shared__ int   idx[N];
  __shared__ float pa[N];
  __shared__ float pb[N];
  int b = blockIdx.x, t = threadIdx.x;
  for (int i = t; i < N; i += 1024) { key[i] = ws[U_OFF + (size_t)b * N + i]; idx[i] = i; }
  __syncthreads();
  for (int k = 2; k <= N; k <<= 1)
    for (int j = k >> 1; j > 0; j >>= 1) {
      for (int i = t; i < N; i += 1024) {
        int x = i ^ j;
        if (x > i) {
          float ka = key[i], kb = key[x];
          int ia = idx[i], ib = idx[x];
          bool up = ((i & k) == 0);
          bool gt = (ka > kb) || (ka == kb && ia > ib); // stable-ascending tiebreak
          if (gt == up) { key[i] = kb; key[x] = ka; idx[i] = ib; idx[x] = ia; }
        }
      }
      __syncthreads();
    }
  // inclusive multiplicative scan (Hillis-Steele, ping-pong)
  float* src = pa; float* dst = pb;
  for (int i = t; i < N; i += 1024) pa[i] = key[i];
  __syncthreads();
  for (int off = 1; off < N; off <<= 1) {
    for (int i = t; i < N; i += 1024)
      dst[i] = src[i] * ((i >= off) ? src[i - off] : 1.f);
    __syncthreads();
    float* tmp = src; src = dst; dst = tmp;
  }
  for (int i = t; i < N; i += 1024) {
    float excl = (i == 0) ? 1.f : src[i - 1];
    ws[ALLOC_OFF + (size_t)b * N + idx[i]] = (1.f - key[i]) * excl;
  }
}

// ---------- K3: write softmax, ww, padded rw-stack, Swr (one block per batch) ----------
__global__ void k3_ww(const float* __restrict__ rw_in, const float* __restrict__ agate,
                      const float* __restrict__ wgate, float* __restrict__ ws) {
  __shared__ float red[256];
  int b = blockIdx.x, t = threadIdx.x;
  float x[8]; float mx = -1e30f;
  for (int k = 0; k < 8; ++k) {
    int n = t + 256 * k;
    x[k] = ws[LOGW_OFF + (size_t)b * N + n];
    mx = fmaxf(mx, x[k]);
  }
  float m = blk_max(red, t, 256, mx);
  float se = 0.f;
  for (int k = 0; k < 8; ++k) se += __expf(x[k] - m);
  float s = blk_sum(red, t, 256, se);
  float ag = agate[b], wg = wgate[b];
  float swr[R] = {0.f, 0.f, 0.f, 0.f};
  for (int k = 0; k < 8; ++k) {
    int n = t + 256 * k;
    size_t id = (size_t)b * N + n;
    float cwv = __expf(x[k] - m) / s;
    float a = ws[ALLOC_OFF + id];
    float wwv = wg * (ag * a + (1.f - ag) * cwv);
    ws[WW_OFF + id] = wwv;
    for (int r = 0; r < R; ++r) {
      float rv = rw_in[id * R + r];
      ws[RWSTK_OFF + id * 16 + r] = rv;
      ws[RWSTK_OFF + id * 16 + 4 + r] = (1.f - wwv) * rv;
      ws[RWSTK_OFF + id * 16 + 8 + r] = 0.f;
      ws[RWSTK_OFF + id * 16 + 12 + r] = 0.f;
      swr[r] += wwv * rv;
    }
  }
  for (int r = 0; r < R; ++r) {
    float tot = blk_sum(red, t, 256, swr[r]);
    if (t == 0) ws[SCAL_OFF + SWR_S + b * R + r] = tot;
  }
}

// ---------- K4: mem_new (erase+add) fused with read-key cosine logits ----------
__global__ void k4_memnew_phir(const float* __restrict__ mem, const float* __restrict__ rk,
                               const float* __restrict__ rstr, const float* __restrict__ wv,
                               const float* __restrict__ ev, float* __restrict__ ws) {
  int wid = threadIdx.x >> 5, lane = threadIdx.x & 31;
  int g = blockIdx.x * 8 + wid;
  int b = g >> 11;
  size_t row = (size_t)g * C;
  float wwv = ws[WW_OFF + g];
  float nrm = 0.f, d0 = 0.f, d1 = 0.f, d2 = 0.f, d3 = 0.f;
  const float4* rk4 = (const float4*)(rk + (size_t)b * C * R);
  for (int it = 0; it < C / 32; ++it) {
    int c = lane + 32 * it;
    float m = mem[row + c];
    float mn = m * (1.f - wwv * ev[b * C + c]) + wwv * wv[b * C + c];
    ws[MEMNEW_OFF + row + c] = mn;
    nrm += mn * mn;
    float4 kv = rk4[c];
    d0 += mn * kv.x; d1 += mn * kv.y; d2 += mn * kv.z; d3 += mn * kv.w;
  }
  for (int off = 16; off > 0; off >>= 1) {
    nrm += __shfl_xor(nrm, off, 32);
    d0 += __shfl_xor(d0, off, 32); d1 += __shfl_xor(d1, off, 32);
    d2 += __shfl_xor(d2, off, 32); d3 += __shfl_xor(d3, off, 32);
  }
  if (lane == 0) {
    float mnn = sqrtf(nrm);
    float dd[R] = {d0, d1, d2, d3};
    for (int r = 0; r < R; ++r) {
      float kn = ws[SCAL_OFF + RKN_S + b * R + r];
      ws[LOGR_OFF + (size_t)g * R + r] = dd[r] / (mnn * kn + EPS) * rstr[b * R + r];
    }
  }
}

// ---------- K5: read softmax over N per (b,r) column ----------
__global__ void k5_softmax_r(float* __restrict__ ws) {
  __shared__ float red[256];
  int b = blockIdx.x >> 2, r = blockIdx.x & 3, t = threadIdx.x;
  float x[8]; float mx = -1e30f;
  for (int k = 0; k < 8; ++k) {
    int n = t + 256 * k;
    x[k] = ws[LOGR_OFF + ((size_t)b * N + n) * R + r];
    mx = fmaxf(mx, x[k]);
  }
  float m = blk_max(red, t, 256, mx);
  float se = 0.f;
  for (int k = 0; k < 8; ++k) se += __expf(x[k] - m);
  float s = blk_sum(red, t, 256, se);
  for (int k = 0; k < 8; ++k) {
    int n = t + 256 * k;
    ws[LOGR_OFF + ((size_t)b * N + n) * R + r] = __expf(x[k] - m) / s;
  }
}

// ---------- K6: fwd link pass, WMMA f32 16x16x4 (L @ [rw|rw1|0...]) ----------
__global__ __launch_bounds__(128) void k6_link_fwd(const float* __restrict__ link,
                                                   float* __restrict__ ws) {
  const int b = blockIdx.y;
  const int i0 = blockIdx.x * 16;
  const int wave = threadIdx.x >> 5;
  const int lane = threadIdx.x & 31;
  const int lo = lane & 15, hi = lane >> 4;
  const float* rwst = ws + RWSTK_OFF + (size_t)b * N * 16 + lo;
  v8f c = {};
  const size_t lbase = (size_t)b * N * N + (size_t)(i0 + lo) * N;
  const int k0 = wave * (N / 4);
#pragma unroll 4
  for (int k = k0; k < k0 + N / 4; k += 4) {
    const int kb = k + 2 * hi;
    const float2 av = *(const float2*)(link + lbase + kb);
    v2f a; a.x = av.x; a.y = av.y;
    v2f bb;
    bb.x = rwst[(size_t)kb * 16];
    bb.y = rwst[(size_t)(kb + 1) * 16];
    c = __builtin_amdgcn_wmma_f32_16x16x4_f32(false, a, false, bb, (short)0, c, false, false);
  }
  __shared__ float red[4][32][8];
  for (int v = 0; v < 8; ++v) red[wave][lane][v] = c[v];
  __syncthreads();
  if (wave == 0) {
    for (int v = 0; v < 8; ++v) {
      if (lo < 8) {
        float s = red[0][lane][v] + red[1][lane][v] + red[2][lane][v] + red[3][lane][v];
        int i = i0 + v + 8 * hi;
        ws[FWD2_OFF + ((size_t)b * N + i) * 8 + lo] = s;
      }
    }
  }
}

// ---------- K7: bwd link pass, WMMA f32 16x16x4 (L^T @ [rw|rw1|0...]) ----------
__global__ __launch_bounds__(128) void k7_link_bwd(const float* __restrict__ link,
                                                   float* __restrict__ ws) {
  const int b = blockIdx.y;
  const int i0 = blockIdx.x * 16;
  const int wave = threadIdx.x >> 5;
  const int lane = threadIdx.x & 31;
  const int lo = lane & 15, hi = lane >> 4;
  const float* rwst = ws + RWSTK_OFF + (size_t)b * N * 16 + lo;
  v8f c = {};
  const size_t cb = (size_t)b * N * N + i0 + lo;
  const int k0 = wave * (N / 4);
#pragma unroll 4
  for (int k = k0; k < k0 + N / 4; k += 4) {
    const int kr = k + 2 * hi;
    v2f a;
    a.x = link[cb + (size_t)kr * N];
    a.y = link[cb + (size_t)(kr + 1) * N];
    v2f bb;
    bb.x = rwst[(size_t)kr * 16];
    bb.y = rwst[(size_t)(kr + 1) * 16];
    c = __builtin_amdgcn_wmma_f32_16x16x4_f32(false, a, false, bb, (short)0, c, false, false);
  }
  __shared__ float red[4][32][8];
  for (int v = 0; v < 8; ++v) red[wave][lane][v] = c[v];
  __syncthreads();
  if (wave == 0) {
    for (int v = 0; v < 8; ++v) {
      if (lo < 8) {
        float s = red[0][lane][v] + red[1][lane][v] + red[2][lane][v] + red[3][lane][v];
        int i = i0 + v + 8 * hi;
        ws[BWD2_OFF + ((size_t)b * N + i) * 8 + lo] = s;
      }
    }
  }
}

// ---------- K8: assemble fwd/bwd (rank-1 + diag corrections) and rw_new ----------
__global__ void k8_combine(const float* __restrict__ link, const float* __restrict__ prec,
                           const float* __restrict__ rw_in, const float* __restrict__ rmodes,
                           float* __restrict__ ws) {
  int id = blockIdx.x * 256 + threadIdx.x;  // b*N+n
  int b = id >> 11, n = id & (N - 1);
  float wwv = ws[WW_OFF + id];
  float pv = prec[id];
  float ld = link[(size_t)b * N * N + (size_t)n * N + n];
  for (int r = 0; r < R; ++r) {
    float rv = rw_in[(size_t)id * R + r];
    float diag = ((1.f - 2.f * wwv) * ld + wwv * pv) * rv;
    float fB = ws[FWD2_OFF + (size_t)id * 8 + r];      // L @ rw
    float fA = ws[FWD2_OFF + (size_t)id * 8 + 4 + r];  // L @ rw1
    float bB = ws[BWD2_OFF + (size_t)id * 8 + r];      // L^T @ rw
    float bA = ws[BWD2_OFF + (size_t)id * 8 + 4 + r];  // L^T @ rw1
    float spr = ws[SCAL_OFF + SPR_S + b * R + r];
    float swr = ws[SCAL_OFF + SWR_S + b * R + r];
    float fwd = fA - wwv * fB + wwv * spr - diag;
    float bwd = bA - wwv * bB + pv * swr - diag;
    float crv = ws[LOGR_OFF + (size_t)id * R + r];
    float m0 = rmodes[b * 3 * R + r];
    float m1 = rmodes[b * 3 * R + R + r];
    float m2 = rmodes[b * 3 * R + 2 * R + r];
    ws[RWNEW_OFF + (size_t)id * R + r] = m0 * bwd + m1 * crv + m2 * fwd;
  }
}

// ---------- K9: read_vectors[b,c,r] = sum_n mem_new * rw_new (split-K atomics) ----------
__global__ void k9_readvec(float* __restrict__ out, const float* __restrict__ ws) {
  int b = blockIdx.x >> 3, s = blockIdx.x & 7;
  int c = threadIdx.x;
  float acc[R] = {0.f, 0.f, 0.f, 0.f};
  for (int n = s * 256; n < s * 256 + 256; ++n) {
    size_t id = (size_t)b * N + n;
    float m = ws[MEMNEW_OFF + id * C + c];
    const float4 rv = *(const float4*)(ws + RWNEW_OFF + id * R);
    acc[0] += m * rv.x; acc[1] += m * rv.y; acc[2] += m * rv.z; acc[3] += m * rv.w;
  }
  for (int r = 0; r < R; ++r)
    atomicAdd(&out[((size_t)b * C + c) * R + r], acc[r]);
}

extern "C" void kernel_launch(void* const* d_in, const int* in_sizes, int n_in,
                              void* d_out, int out_size, void* d_ws, size_t ws_size,
                              hipStream_t stream) {
  const float* mem    = (const float*)d_in[0];
  const float* link   = (const float*)d_in[1];
  const float* usage  = (const float*)d_in[2];
  const float* rw_in  = (const float*)d_in[3];
  const float* wwp    = (const float*)d_in[4];
  const float* prec   = (const float*)d_in[5];
  const float* rk     = (const float*)d_in[6];
  const float* rstr   = (const float*)d_in[7];
  const float* wk     = (const float*)d_in[8];
  const float* wstr   = (const float*)d_in[9];
  const float* fg     = (const float*)d_in[10];
  const float* agate  = (const float*)d_in[11];
  const float* wgate  = (const float*)d_in[12];
  const float* wv     = (const float*)d_in[13];
  const float* ev     = (const float*)d_in[14];
  const float* rmodes = (const float*)d_in[15];
  float* ws = (float*)d_ws;
  float* out = (float*)d_out;

  k0_scalars<<<B, 256, 0, stream>>>(wk, rk, prec, rw_in, ws);
  k1_usage_phiw<<<(B * N) / 8, 256, 0, stream>>>(mem, usage, rw_in, wwp, fg, wk, wstr, ws);
  k2_sort_alloc<<<B, 1024, 0, stream>>>(ws);
  k3_ww<<<B, 256, 0, stream>>>(rw_in, agate, wgate, ws);
  k4_memnew_phir<<<(B * N) / 8, 256, 0, stream>>>(mem, rk, rstr, wv, ev, ws);
  k5_softmax_r<<<B * R, 256, 0, stream>>>(ws);
  k6_link_fwd<<<dim3(N / 16, B), 128, 0, stream>>>(link, ws);
  k7_link_bwd<<<dim3(N / 16, B), 128, 0, stream>>>(link, ws);
  k8_combine<<<(B * N) / 256, 256, 0, stream>>>(link, prec, rw_in, rmodes, ws);
  hipMemsetAsync(d_out, 0, (size_t)out_size * sizeof(float), stream);
  k9_readvec<<<B * 8, 256, 0, stream>>>(out, ws);
}